// QAGATConv_72327249264854
// MI455X (gfx1250) — compile-verified
//
#include <hip/hip_runtime.h>
#include <hip/hip_bf16.h>

// GATConv on MI455X (gfx1250, wave32):
//   ft = feat @ fc_w.T         -> f32 WMMA 16x16x4 (exact f32, GEMM is BW-bound anyway)
//   el/er                      -> elementwise reduction kernel
//   segment max/sum passes     -> L2 atomics (uint-key float max, f32 atomic add)
//   rst[dst] += ft[src]*c      -> scatter atomic adds (dominant: ~0.65 GB L2 traffic)

typedef float v2f __attribute__((ext_vector_type(2)));
typedef float v8f __attribute__((ext_vector_type(8)));

#define NEG_SLOPE 0.2f

__device__ __forceinline__ float lrelu(float x) {
    return x > 0.0f ? x : NEG_SLOPE * x;
}

// Monotone order-preserving float<->uint mapping for atomicMax on signed floats.
__device__ __forceinline__ unsigned f2key(float f) {
    unsigned u = __float_as_uint(f);
    return (u & 0x80000000u) ? ~u : (u | 0x80000000u);
}
__device__ __forceinline__ float key2f(unsigned k) {
    unsigned u = (k & 0x80000000u) ? (k ^ 0x80000000u) : ~k;
    return __uint_as_float(u);
}

// ---------------- init: rst = bias broadcast, zero accumulators ----------------
__global__ void k_init(float* __restrict__ rst, const float* __restrict__ bias,
                       unsigned* __restrict__ emax, float* __restrict__ esum,
                       float* __restrict__ csum, float* __restrict__ prsum, int N) {
    int i = blockIdx.x * blockDim.x + threadIdx.x;
    if (i < N * 128) rst[i] = bias[i & 127];
    if (i < N * 4) { emax[i] = 0u; esum[i] = 0.0f; csum[i] = 0.0f; }
    if (i < N) prsum[i] = 0.0f;
}

// ---------------- GEMM: ft[N,128] = feat[N,256] @ fc_w[128,256]^T --------------
// One wave per 16-row stripe; accumulates all 8 column tiles (N=128) so each
// A-fragment feeds 8 WMMAs. 64 k-steps of V_WMMA_F32_16X16X4_F32.
__global__ __launch_bounds__(32) void k_gemm_wmma(
    const float* __restrict__ feat,   // [N, 256]
    const float* __restrict__ fcw,    // [128, 256]  (B[k][n] = fcw[n*256 + k])
    float* __restrict__ ft)           // [N, 128]
{
    const int row0  = blockIdx.x * 16;
    const int lane  = threadIdx.x;
    const int lh    = lane & 15;
    const int khalf = (lane >> 4) << 1;   // 0 for lanes 0-15, 2 for lanes 16-31

    v8f acc[8];
    #pragma unroll
    for (int t = 0; t < 8; ++t) acc[t] = (v8f){0,0,0,0,0,0,0,0};

    const float* arow = feat + (size_t)(row0 + lh) * 256;

    for (int kk = 0; kk < 256; kk += 4) {
        const int k0 = kk + khalf;
        // A fragment 16x4 f32: lanes 0-15 -> K=kk,kk+1 ; lanes 16-31 -> K=kk+2,kk+3
        v2f a;
        a.x = arow[k0];
        a.y = arow[k0 + 1];
        #pragma unroll
        for (int t = 0; t < 8; ++t) {
            // B fragment 4x16 f32, lane holds column n = t*16+lh, same K split as A
            const float* brow = fcw + (size_t)(t * 16 + lh) * 256;
            v2f b;
            b.x = brow[k0];
            b.y = brow[k0 + 1];
            acc[t] = __builtin_amdgcn_wmma_f32_16x16x4_f32(
                false, a, false, b, (short)0, acc[t], false, false);
        }
    }

    // C/D layout: VGPR r -> M = r + 8*(lane>=16), N = lane&15
    const int mbase = row0 + ((lane >> 4) << 3);
    #pragma unroll
    for (int t = 0; t < 8; ++t) {
        #pragma unroll
        for (int r = 0; r < 8; ++r) {
            ft[(size_t)(mbase + r) * 128 + t * 16 + lh] = acc[t][r];
        }
    }
}

// ---------------- el/er: per-(node,head) dot with attn vectors -----------------
__global__ void k_elr(const float* __restrict__ ft, const float* __restrict__ al,
                      const float* __restrict__ ar, float* __restrict__ el,
                      float* __restrict__ er, int NH) {
    int i = blockIdx.x * blockDim.x + threadIdx.x;
    if (i >= NH) return;
    int h = i & 3;
    const float* f  = ft + (size_t)i * 32;
    const float* pl = al + h * 32;
    const float* pr = ar + h * 32;
    float sl = 0.0f, sr = 0.0f;
    #pragma unroll
    for (int d = 0; d < 32; ++d) {
        float v = f[d];
        sl += v * pl[d];
        sr += v * pr[d];
    }
    el[i] = sl;
    er[i] = sr;
}

// ---------------- pass 1: segment max over incoming edges ----------------------
__global__ void k_edge_max(const int* __restrict__ src, const int* __restrict__ dst,
                           const float* __restrict__ el, const float* __restrict__ er,
                           unsigned* __restrict__ emax, int E) {
    int e = blockIdx.x * blockDim.x + threadIdx.x;
    if (e >= E) return;
    int s = src[e], d = dst[e];
    #pragma unroll
    for (int h = 0; h < 4; ++h) {
        float x = lrelu(el[s * 4 + h] + er[d * 4 + h]);
        atomicMax(&emax[d * 4 + h], f2key(x));
    }
}

// ---------------- pass 2: exp-sum + pagerank-sum -------------------------------
__global__ void k_edge_expsum(const int* __restrict__ src, const int* __restrict__ dst,
                              const float* __restrict__ pr, const float* __restrict__ el,
                              const float* __restrict__ er, const unsigned* __restrict__ emax,
                              float* __restrict__ esum, float* __restrict__ prsum, int E) {
    int e = blockIdx.x * blockDim.x + threadIdx.x;
    if (e >= E) return;
    int s = src[e], d = dst[e];
    atomicAdd(&prsum[d], pr[s]);
    #pragma unroll
    for (int h = 0; h < 4; ++h) {
        float x = lrelu(el[s * 4 + h] + er[d * 4 + h]);
        float m = key2f(emax[d * 4 + h]);
        atomicAdd(&esum[d * 4 + h], __expf(x - m));
    }
}

// ---------------- pass 3: c = softmax * (pr/prsum); csum -----------------------
__global__ void k_edge_csum(const int* __restrict__ src, const int* __restrict__ dst,
                            const float* __restrict__ pr, const float* __restrict__ el,
                            const float* __restrict__ er, const unsigned* __restrict__ emax,
                            const float* __restrict__ esum, const float* __restrict__ prsum,
                            float* __restrict__ csum, int E) {
    int e = blockIdx.x * blockDim.x + threadIdx.x;
    if (e >= E) return;
    int s = src[e], d = dst[e];
    float b = pr[s] / prsum[d];
    #pragma unroll
    for (int h = 0; h < 4; ++h) {
        float x  = lrelu(el[s * 4 + h] + er[d * 4 + h]);
        float m  = key2f(emax[d * 4 + h]);
        float a  = __expf(x - m) / esum[d * 4 + h];
        atomicAdd(&csum[d * 4 + h], a * b);
    }
}

// ---------------- pass 4: rst[dst] += ft[src] * (c/csum) -----------------------
// One thread per (edge, head); 32 gathered reads + 32 scatter atomic adds.
__global__ void k_edge_aggr(const int* __restrict__ src, const int* __restrict__ dst,
                            const float* __restrict__ pr, const float* __restrict__ el,
                            const float* __restrict__ er, const unsigned* __restrict__ emax,
                            const float* __restrict__ esum, const float* __restrict__ prsum,
                            const float* __restrict__ csum, const float* __restrict__ ft,
                            float* __restrict__ rst, int EH) {
    int idx = blockIdx.x * blockDim.x + threadIdx.x;
    if (idx >= EH) return;
    int e = idx >> 2, h = idx & 3;
    int s = src[e], d = dst[e];
    float x  = lrelu(el[s * 4 + h] + er[d * 4 + h]);
    float m  = key2f(emax[d * 4 + h]);
    float a  = __expf(x - m) / esum[d * 4 + h];
    float b  = pr[s] / prsum[d];
    float cf = (a * b) / csum[d * 4 + h];

    const float* fsrc = ft + (size_t)s * 128 + h * 32;
    float* rdst       = rst + (size_t)d * 128 + h * 32;
    #pragma unroll
    for (int dd = 0; dd < 32; ++dd) {
        atomicAdd(&rdst[dd], fsrc[dd] * cf);
    }
}

extern "C" void kernel_launch(void* const* d_in, const int* in_sizes, int n_in,
                              void* d_out, int out_size, void* d_ws, size_t ws_size,
                              hipStream_t stream) {
    const float* feat   = (const float*)d_in[0];
    const float* pr     = (const float*)d_in[1];
    const int*   esrc   = (const int*)d_in[2];
    const int*   edst   = (const int*)d_in[3];
    const float* fcw    = (const float*)d_in[4];
    const float* attn_l = (const float*)d_in[5];
    const float* attn_r = (const float*)d_in[6];
    const float* bias   = (const float*)d_in[7];
    float* rst = (float*)d_out;

    const int N = in_sizes[1];   // 20000 (multiple of 16)
    const int E = in_sizes[2];   // 640000

    // carve workspace (~11.9 MB)
    char* ws = (char*)d_ws;
    size_t off = 0;
    auto carve = [&](size_t bytes) -> char* {
        char* p = ws + off;
        off += (bytes + 255) & ~(size_t)255;
        return p;
    };
    float*    ft    = (float*)carve((size_t)N * 128 * sizeof(float));
    float*    el    = (float*)carve((size_t)N * 4 * sizeof(float));
    float*    er    = (float*)carve((size_t)N * 4 * sizeof(float));
    unsigned* emax  = (unsigned*)carve((size_t)N * 4 * sizeof(unsigned));
    float*    esum  = (float*)carve((size_t)N * 4 * sizeof(float));
    float*    csum  = (float*)carve((size_t)N * 4 * sizeof(float));
    float*    prsum = (float*)carve((size_t)N * sizeof(float));

    const int B = 256;
    k_init<<<(N * 128 + B - 1) / B, B, 0, stream>>>(rst, bias, emax, esum, csum, prsum, N);
    k_gemm_wmma<<<N / 16, 32, 0, stream>>>(feat, fcw, ft);
    k_elr<<<(N * 4 + B - 1) / B, B, 0, stream>>>(ft, attn_l, attn_r, el, er, N * 4);
    k_edge_max<<<(E + B - 1) / B, B, 0, stream>>>(esrc, edst, el, er, emax, E);
    k_edge_expsum<<<(E + B - 1) / B, B, 0, stream>>>(esrc, edst, pr, el, er, emax, esum, prsum, E);
    k_edge_csum<<<(E + B - 1) / B, B, 0, stream>>>(esrc, edst, pr, el, er, emax, esum, prsum, csum, E);
    k_edge_aggr<<<(E * 4 + B - 1) / B, B, 0, stream>>>(esrc, edst, pr, el, er, emax, esum, prsum,
                                                       csum, ft, rst, E * 4);
}